// Convolution_35373350650127
// MI455X (gfx1250) — compile-verified
//
#include <hip/hip_runtime.h>
#include <math.h>

typedef float v2f __attribute__((ext_vector_type(2)));
typedef float v8f __attribute__((ext_vector_type(8)));
typedef unsigned uv4 __attribute__((ext_vector_type(4)));
typedef int iv4 __attribute__((ext_vector_type(4)));
typedef int iv8 __attribute__((ext_vector_type(8)));

#define TZ 4
#define TY 8
#define TX 16
#define HZ (TZ + 4)            // 8
#define HY (TY + 4)            // 12
#define HX (TX + 4)            // 20
#define NSP (HZ * HY * HX)     // 1920 halo voxels per channel
#define NCH 8                  // input channels per K-chunk
#define XPITCH 12              // LDS pitch per voxel (8 ch + 4 pad dwords)
#define XBUF (NSP * XPITCH)    // 23040 dwords per x buffer
#define WROW 10                // padded W row pitch (dwords): 8 data + 2 pad
#define WNROW 1600             // 25 taps * 64 o rows per slab
#define WBUF (WNROW * WROW)    // 16000 dwords per W buffer

#if defined(__has_builtin)
#if __has_builtin(__builtin_amdgcn_tensor_load_to_lds)
#define HAVE_TDM 1
#endif
#endif
#ifndef HAVE_TDM
#define HAVE_TDM 0
#endif

#if HAVE_TDM
// Issue one TENSOR_LOAD_TO_LDS. Descriptor packing per CDNA5 ISA ch.8.
static __device__ __forceinline__ void tdm_load(
    unsigned lds_byte, unsigned long long gaddr, unsigned g1d0,
    unsigned td0, unsigned td1, unsigned tile0, unsigned tile1, unsigned tile2,
    unsigned long long s0, unsigned long long s1, unsigned td2)
{
  uv4 g0;
  g0.x = 1u;                                        // count=1, user mode
  g0.y = lds_byte;                                  // lds_addr
  g0.z = (unsigned)gaddr;                           // global_addr[31:0]
  g0.w = ((unsigned)(gaddr >> 32) & 0x01FFFFFFu) | (2u << 30);  // addr|type=2
  iv8 g1;
  g1[0] = (int)g1d0;                                // wg_mask=0|data_size|pad ctl
  g1[1] = (int)((td0 & 0xFFFFu) << 16);             // abar=0 | tensor_dim0.lo
  g1[2] = (int)((td0 >> 16) | ((td1 & 0xFFFFu) << 16));
  g1[3] = (int)((td1 >> 16) | (tile0 << 16));
  g1[4] = (int)(tile1 | (tile2 << 16));
  g1[5] = (int)(unsigned)s0;                        // dim0_stride[31:0]
  g1[6] = (int)((unsigned)(s0 >> 32) | ((unsigned)(s1 & 0xFFFFu) << 16));
  g1[7] = (int)(unsigned)(s1 >> 16);                // dim1_stride[47:16]
  iv4 g2;
  g2[0] = (int)td2;
  g2[1] = 0;
  g2[2] = 0;
  g2[3] = 0;
  iv4 g3 = {0, 0, 0, 0};
#if __clang_major__ >= 23
  __builtin_amdgcn_tensor_load_to_lds(g0, g1, g2, g3, (iv8)0, 0);
#else
  __builtin_amdgcn_tensor_load_to_lds(g0, g1, g2, g3, 0);
#endif
}
#endif  // HAVE_TDM

// ---------------------------------------------------------------------------
// Kernel 1: build fused conv weights W[tap][o][i] (125 x 64 x 64 fp32, 2 MB).
// ---------------------------------------------------------------------------
__global__ __launch_bounds__(256) void prep_kernel(
    const float* __restrict__ lw0, const float* __restrict__ lw1,
    const float* __restrict__ tpw, float* __restrict__ Wg)
{
  __shared__ float emb[1024];
  const int tap = blockIdx.x;
  const int kd = tap / 25, kh = (tap / 5) % 5, kw = tap % 5;
  const float rx = -1.f + 0.5f * kd, ry = -1.f + 0.5f * kh, rz = -1.f + 0.5f * kw;
  const float d = sqrtf(rx * rx + ry * ry + rz * rz);
  const float invd = d > 0.f ? 1.f / d : 0.f;
  const float sq3 = 1.7320508075688772f;
  const float sh1[3] = { sq3 * ry * invd, sq3 * rz * invd, sq3 * rx * invd };
  float e[5];
#pragma unroll
  for (int j = 0; j < 5; ++j) {
    float t = (d - 0.25f * j) * 4.f;
    e[j] = expf(-t * t) * (1.f / 1.12f);
  }
  const float scale = cosf(3.14159265358979f * d) * (1.f / 11.180339887498949f);
  for (int k = (int)threadIdx.x; k < 1024; k += 256) {
    float s = 0.f;
#pragma unroll
    for (int j = 0; j < 5; ++j) s += e[j] * tpw[j * 1024 + k];
    emb[k] = s * scale;
  }
  __syncthreads();

  const bool center = (tap == 62);                 // kd=kh=kw=2
  const float A0  = 0.17677669529663687f;          // sqrt(1/32)
  const float A0I = 0.10206207261596575f;          // A0/sqrt(3)
  const float A1I = 0.17677669529663687f;          // sqrt(3/32)/sqrt(3) == A0
  for (int idx = (int)threadIdx.x; idx < 4096; idx += 256) {
    const int o = idx >> 6, i = idx & 63;
    float v;
    if (o < 16) {
      if (i < 16) {
        v = 0.1f * A0 * emb[i * 16 + o];
        if (center) v += 0.25f * lw0[i * 16 + o];
      } else {
        const int u = (i - 16) / 3, ic = (i - 16) % 3;
        v = 0.1f * A0I * emb[768 + u * 16 + o] * sh1[ic];
      }
    } else {
      const int w = (o - 16) / 3, kc = (o - 16) % 3;
      if (i < 16) {
        v = 0.1f * A1I * emb[256 + i * 16 + w] * sh1[kc];
      } else {
        const int u = (i - 16) / 3, ic = (i - 16) % 3;
        v = 0.f;
        if (ic == kc) {
          v = 0.1f * A1I * emb[512 + u * 16 + w];
          if (center) v += 0.25f * lw1[u * 16 + w];
        }
      }
    }
    Wg[(size_t)tap * 4096 + idx] = v;
  }
}

// ---------------------------------------------------------------------------
// Kernel 2: implicit-GEMM 5x5x5 conv via V_WMMA_F32_16X16X4_F32.
// 1024 blocks, 512 threads (16 waves). Out tile: 64o x (4z,8y,16x).
// Steady state: ds_load_2addr_b64 + wmma only; x chunks double-buffered via
// async-LDS loads (ASYNCcnt), W slabs double-buffered via TDM (TENSORcnt).
// ---------------------------------------------------------------------------
__global__ __launch_bounds__(512) void conv_kernel(
    const float* __restrict__ X, const float* __restrict__ Wg,
    float* __restrict__ out)
{
  __shared__ float xs[2 * XBUF];       // 184,320 B
  __shared__ float Wlds[2 * WBUF];     // 128,000 B   (total 305 KB <= 320 KB)

  const int id = (int)blockIdx.x;
  const int x0 = (id & 3) * TX;
  const int y0 = ((id >> 2) & 7) * TY;
  const int z0 = ((id >> 5) & 15) * TZ;
  const int b  = id >> 9;
  const int tid = (int)threadIdx.x;
  const int lane = tid & 31;
  const int wv = tid >> 5;
  const int m = wv & 3, zh = wv >> 2;
  const int hi = lane >> 4, ln = lane & 15;
  const int o0 = m * 16;

  v8f acc[8];
#pragma unroll
  for (int j = 0; j < 8; ++j)
    acc[j] = (v8f){0.f, 0.f, 0.f, 0.f, 0.f, 0.f, 0.f, 0.f};

  const float* Xb = X + (size_t)b * 64 * 262144;
  const bool interior = (z0 >= 2) && (z0 + HZ <= 66) &&
                        (y0 >= 2) && (y0 + HY <= 66) &&
                        (x0 >= 2) && (x0 + HX <= 66);

  // Fire-and-forget loader for x chunk qc into buffer qc&1 (layout [sp][12]).
  auto loadx = [&](int qc) {
    float* dst = xs + (qc & 1) * XBUF;
    const int cq = qc * NCH;
    if (interior) {
      for (int idx = tid; idx < NCH * NSP; idx += 512) {   // 30 iters
        const int xx = idx % HX;
        int t = idx / HX;
        const int yy = t % HY; t /= HY;
        const int zz = t & 7;
        const int ch = t >> 3;
        const float* g = Xb + (size_t)(cq + ch) * 262144 +
                         (size_t)(z0 - 2 + zz) * 4096 + (y0 - 2 + yy) * 64 +
                         (x0 - 2 + xx);
        const unsigned loff =
            (unsigned)(size_t)&dst[((zz * HY + yy) * HX + xx) * XPITCH + ch];
        asm volatile("global_load_async_to_lds_b32 %0, %1, off"
                     :: "v"(loff), "v"(g) : "memory");
      }
    } else {
      for (int idx = tid; idx < NCH * NSP; idx += 512) {
        const int xx = idx % HX;
        int t = idx / HX;
        const int yy = t % HY; t /= HY;
        const int zz = t & 7;
        const int ch = t >> 3;
        const int gz = z0 - 2 + zz, gy = y0 - 2 + yy, gx = x0 - 2 + xx;
        float v = 0.f;
        if ((unsigned)gz < 64u && (unsigned)gy < 64u && (unsigned)gx < 64u)
          v = Xb[(size_t)(cq + ch) * 262144 + (size_t)gz * 4096 + gy * 64 + gx];
        dst[((zz * HY + yy) * HX + xx) * XPITCH + ch] = v;
      }
    }
  };

#if HAVE_TDM
  // Issue TDM for W slab s (= q*5 + kd) into buffer s&1; wave 0 only.
  auto loadW = [&](int s) {
    if (wv == 0) {
      const int kd = s % 5, qc = s / 5;
      const unsigned lds = (unsigned)(size_t)&Wlds[(s & 1) * WBUF];
      const unsigned long long ga = (unsigned long long)(size_t)(
          Wg + (size_t)kd * 102400 + qc * NCH);
      tdm_load(lds, ga,
               (2u << 16) | (1u << 20) | (2u << 22) | (1u << 25),
               NCH, WNROW, NCH, WNROW, 0, 64ull, 0ull, 0);
    }
  };
#endif

  // Prologue: chunk 0 of x, slab 0 of W.
  loadx(0);
#if HAVE_TDM
  loadW(0);
#endif
  asm volatile("s_wait_asynccnt 0" ::: "memory");
#if HAVE_TDM
  __builtin_amdgcn_s_wait_tensorcnt(0);
#endif
  __syncthreads();

  for (int q = 0; q < 8; ++q) {
    const float* xb = xs + (q & 1) * XBUF;
    if (q < 7) loadx(q + 1);                 // overlap next x chunk with compute
    for (int kd = 0; kd < 5; ++kd) {
      const int slab = q * 5 + kd;
      const float* Wb = Wlds + (slab & 1) * WBUF;
#if HAVE_TDM
      if (slab < 39) loadW(slab + 1);        // prefetch next W slab
#else
      {  // synchronous cooperative slab load (fallback, no TDM)
        float* Ww = Wlds + (slab & 1) * WBUF;
        const float* gw = Wg + (size_t)kd * 102400 + q * NCH;
        for (int idx = tid; idx < WNROW * NCH; idx += 512) {
          const int col = idx & 7, row = idx >> 3;
          Ww[row * WROW + col] = gw[(size_t)row * 64 + col];
        }
        __syncthreads();
      }
#endif
      const int zz = zh + kd;
#pragma unroll 1
      for (int kk = 0; kk < 25; ++kk) {
        const int kh = kk / 5, kw = kk % 5;
        const float* arow = Wb + (kk * 64 + o0 + ln) * WROW + 2 * hi;
        const float* bb = xb + ((zz * HY + kh) * HX + ln + kw) * XPITCH + 2 * hi;
#pragma unroll
        for (int k4 = 0; k4 < 2; ++k4) {
          // A (16x4): lane(hi,ln) holds W[tap][o0+ln][k4*4 + 2hi + {0,1}]
          const v2f a = *(const v2f*)(arow + k4 * 4);
          const float* bp = bb + k4 * 4;
#pragma unroll
          for (int j = 0; j < 8; ++j) {
            // B (4x16): lane(hi,ln) holds x[ch=k4*4+2hi+{0,1}][n=ln] - one b64
            const v2f bv = *(const v2f*)(bp + j * (HX * XPITCH));
            acc[j] = __builtin_amdgcn_wmma_f32_16x16x4_f32(
                false, a, false, bv, (short)0, acc[j], false, false);
          }
        }
      }
#if HAVE_TDM
      __builtin_amdgcn_s_wait_tensorcnt(0);
#endif
      if (kd == 4) asm volatile("s_wait_asynccnt 0" ::: "memory");
      __syncthreads();
    }
  }

  // C/D layout: vgpr g -> rows M=g (lanes 0-15) and M=g+8 (lanes 16-31), N=ln
  float* ob = out + (size_t)b * 64 * 262144;
#pragma unroll
  for (int j = 0; j < 8; ++j) {
#pragma unroll
    for (int g = 0; g < 8; ++g) {
      const int o = o0 + g + 8 * hi;
      ob[(size_t)o * 262144 + (size_t)(z0 + zh) * 4096 + (y0 + j) * 64 +
         (x0 + ln)] = acc[j][g];
    }
  }
}

extern "C" void kernel_launch(void* const* d_in, const int* in_sizes, int n_in,
                              void* d_out, int out_size, void* d_ws,
                              size_t ws_size, hipStream_t stream) {
  (void)in_sizes; (void)n_in; (void)out_size; (void)ws_size;
  const float* x   = (const float*)d_in[0];
  const float* lw0 = (const float*)d_in[1];
  const float* lw1 = (const float*)d_in[2];
  const float* tpw = (const float*)d_in[3];
  float* Wg = (float*)d_ws;  // needs 125*64*64*4 = 2 MB of scratch
  prep_kernel<<<125, 256, 0, stream>>>(lw0, lw1, tpw, Wg);
  conv_kernel<<<1024, 512, 0, stream>>>(x, Wg, (float*)d_out);
}